// HebbianConv2d_26989574488852
// MI455X (gfx1250) — compile-verified
//
#include <hip/hip_runtime.h>
#include <hip/hip_bf16.h>
#include <math.h>

// ---------------- problem constants ----------------
constexpr int BATCH = 16;
constexpr int CIN   = 16;
constexpr int HH    = 256;
constexpr int WW    = 256;
constexpr int COUT  = 32;
constexpr int KH    = 3;
constexpr int KW    = 3;
constexpr int HO    = 254;
constexpr int WO    = 254;
constexpr int L     = HO * WO;          // 64516 positions per image
constexpr int P     = BATCH * L;        // 1032256 total positions
constexpr int KDIM  = CIN * KH * KW;    // 144

typedef __attribute__((ext_vector_type(2))) float v2f;
typedef __attribute__((ext_vector_type(8))) float v8f;

// ws layout (floats)
//  [0,4608)        wT[k][o]  normalized weight, transposed
//  [4608,4640)     S1[o] = sum_p r
//  [4640,4672)     S2[o] = sum_p r^2
//  [4672,9280)     accG[o][k] = sum_p cr[o,p]*patch[p,k]
constexpr int WS_WT  = 0;
constexpr int WS_S1  = 4608;
constexpr int WS_S2  = 4640;
constexpr int WS_ACC = 4672;

// ---------------- kernel 0: weight norm + zero accumulators ----------------
__global__ __launch_bounds__(64) void k_prep(const float* __restrict__ w,
                                             float* __restrict__ ws) {
    const int o = blockIdx.x;      // 32 blocks, one per out channel
    const int t = threadIdx.x;     // 64 threads
    __shared__ float red[64];
    float s = 0.f;
    for (int k = t; k < KDIM; k += 64) { float v = w[o * KDIM + k]; s += v * v; }
    red[t] = s;
    __syncthreads();
    for (int off = 32; off > 0; off >>= 1) {
        if (t < off) red[t] += red[t + off];
        __syncthreads();
    }
    float nrm = sqrtf(red[0]);
    if (nrm == 0.f) nrm = 1.f;
    const float inv = 1.f / nrm;
    for (int k = t; k < KDIM; k += 64) {
        ws[WS_WT + k * COUT + o]  = w[o * KDIM + k] * inv;  // transposed, normalized
        ws[WS_ACC + o * KDIM + k] = 0.f;
    }
    if (t == 0) { ws[WS_S1 + o] = 0.f; ws[WS_S2 + o] = 0.f; }
}

// ---------------- kernel 1: forward conv via WMMA f32 16x16x4 ----------------
// wave tile: 16 positions (M) x 32 channels (N), K = 144 in 36 steps of 4
__global__ __launch_bounds__(256) void k_fwd(const float* __restrict__ x,
                                             const float* __restrict__ ws,
                                             const float* __restrict__ bias,
                                             float* __restrict__ y) {
    __shared__ float sm[8][COUT * 16];       // per-wave D transpose buffer
    const int lane = threadIdx.x & 31;
    const int wid  = threadIdx.x >> 5;
    const int tile = blockIdx.x * 8 + wid;
    constexpr int NT = P / 16;               // 64516 tiles, exact
    if (tile >= NT) return;                  // whole-wave exit; no block barriers used

    const int m  = lane & 15;                // A row  = position within tile
    const int kg = lane >> 4;                // K half
    const int n  = m;                        // B col  = channel (0..15) / +16
    const int p  = tile * 16 + m;
    const int b  = p / L;
    const int l  = p - b * L;
    const int oh = l / WO;
    const int ow = l - oh * WO;
    const float* xb = x + ((size_t)b * CIN) * HH * WW + oh * WW + ow;
    const float* wT = ws + WS_WT;

    v8f acc0 = {};
    v8f acc1 = {};
    for (int k0 = 0; k0 < KDIM; k0 += 4) {
        const int kk  = k0 + kg * 2;         // lane's K rows: kk, kk+1
        const int c0 = kk / 9;        int r0 = kk - c0 * 9;
        const int i0 = r0 / 3;        const int j0 = r0 - i0 * 3;
        const int kk1 = kk + 1;
        const int c1 = kk1 / 9;       int r1 = kk1 - c1 * 9;
        const int i1 = r1 / 3;        const int j1 = r1 - i1 * 3;
        v2f a;
        a.x = xb[(c0 * HH + i0) * WW + j0];
        a.y = xb[(c1 * HH + i1) * WW + j1];
        v2f b0, b1;
        b0.x = wT[kk * COUT + n];        b0.y = wT[kk1 * COUT + n];
        b1.x = wT[kk * COUT + n + 16];   b1.y = wT[kk1 * COUT + n + 16];
        acc0 = __builtin_amdgcn_wmma_f32_16x16x4_f32(false, a, false, b0, (short)0, acc0, false, false);
        acc1 = __builtin_amdgcn_wmma_f32_16x16x4_f32(false, a, false, b1, (short)0, acc1, false, false);
    }

    // D layout: lane holds channel n, VGPR r holds positions r and r+8 (by lane half)
    const float bi0 = bias[n];
    const float bi1 = bias[n + 16];
    float* s = sm[wid];
#pragma unroll
    for (int r = 0; r < 8; ++r) {
        const int mr = r + kg * 8;           // position row
        s[n * 16 + mr]        = acc0[r] + bi0;
        s[(n + 16) * 16 + mr] = acc1[r] + bi1;
    }
    __builtin_amdgcn_wave_barrier();
    asm volatile("s_wait_dscnt 0" ::: "memory");   // per-wave LDS RAW fence

    // coalesced store: 2 channels per iteration, 16 contiguous positions per half-wave
#pragma unroll
    for (int it = 0; it < 16; ++it) {
        const int ch = it * 2 + kg;
        y[((size_t)b * COUT + ch) * L + l] = s[ch * 16 + m];
    }
}

// ---------------- kernel 2: softmax stats  S1 = sum r, S2 = sum r^2 ----------------
__global__ __launch_bounds__(256) void k_rsum(const float* __restrict__ y,
                                              float* __restrict__ ws) {
    __shared__ float s1[COUT], s2[COUT];
    if (threadIdx.x < COUT) { s1[threadIdx.x] = 0.f; s2[threadIdx.x] = 0.f; }
    __syncthreads();

    float l1[COUT], l2[COUT];
#pragma unroll
    for (int c = 0; c < COUT; ++c) { l1[c] = 0.f; l2[c] = 0.f; }

    for (int p = blockIdx.x * 256 + threadIdx.x; p < P; p += gridDim.x * 256) {
        const int b = p / L;
        const int l = p - b * L;
        const float* yp = y + ((size_t)b * COUT) * L + l;
        float v[COUT];
        float mx = -3.402823466e38f;
#pragma unroll
        for (int c = 0; c < COUT; ++c) { v[c] = yp[c * L]; mx = fmaxf(mx, v[c]); }
        float sum = 0.f;
#pragma unroll
        for (int c = 0; c < COUT; ++c) { v[c] = expf(v[c] - mx); sum += v[c]; }
        const float inv = 1.f / sum;
#pragma unroll
        for (int c = 0; c < COUT; ++c) {
            const float r = v[c] * inv;
            l1[c] += r;
            l2[c] += r * r;
        }
    }
#pragma unroll
    for (int c = 0; c < COUT; ++c) { atomicAdd(&s1[c], l1[c]); atomicAdd(&s2[c], l2[c]); }
    __syncthreads();
    if (threadIdx.x < COUT) {
        atomicAdd(&ws[WS_S1 + threadIdx.x], s1[threadIdx.x]);
        atomicAdd(&ws[WS_S2 + threadIdx.x], s2[threadIdx.x]);
    }
}

// ---------------- kernel 3: delta GEMM  acc[o][k] += cr[o,p]*patch[p,k] via WMMA ----------------
// wave chunk: 32 positions; acc = 2(ch tiles) x 9(k tiles) of 16x16 f32
__global__ __launch_bounds__(256) void k_delta(const float* __restrict__ x,
                                               const float* __restrict__ y,
                                               float* __restrict__ ws) {
    __shared__ float crs[8][32 * 32];        // per-wave cr staging [pos_local][ch]
    const int lane = threadIdx.x & 31;
    const int wid  = threadIdx.x >> 5;
    float* cw = crs[wid];

    float invr[COUT];
#pragma unroll
    for (int c = 0; c < COUT; ++c) {
        float s = ws[WS_S1 + c];
        if (s == 0.f) s = 1.f;
        invr[c] = 1.f / s;
    }

    const int m  = lane & 15;
    const int kg = lane >> 4;

    v8f zero = {};
    v8f acc[2][9];
#pragma unroll
    for (int t = 0; t < 2; ++t)
#pragma unroll
        for (int kt = 0; kt < 9; ++kt) acc[t][kt] = zero;

    constexpr int NCH = P / 32;              // 32258, exact
    const int waves = (gridDim.x * blockDim.x) >> 5;
    const int gw    = (blockIdx.x * blockDim.x + threadIdx.x) >> 5;

    for (int chunk = gw; chunk < NCH; chunk += waves) {
        // one position per lane: softmax over 32 channels, cr = r^2 / r_sum
        {
            const int p = chunk * 32 + lane;
            const int b = p / L;
            const int l = p - b * L;
            const float* yp = y + ((size_t)b * COUT) * L + l;
            float v[COUT];
            float mx = -3.402823466e38f;
#pragma unroll
            for (int c = 0; c < COUT; ++c) { v[c] = yp[c * L]; mx = fmaxf(mx, v[c]); }
            float sum = 0.f;
#pragma unroll
            for (int c = 0; c < COUT; ++c) { v[c] = expf(v[c] - mx); sum += v[c]; }
            const float inv = 1.f / sum;
#pragma unroll
            for (int c = 0; c < COUT; ++c) {
                const float r = v[c] * inv;
                cw[lane * 32 + c] = r * r * invr[c];
            }
        }
        __builtin_amdgcn_wave_barrier();
        asm volatile("s_wait_dscnt 0" ::: "memory");

        for (int s = 0; s < 8; ++s) {
            const int pl = s * 4 + kg * 2;   // lane's two K(position) rows
            v2f a0, a1;
            a0.x = cw[pl * 32 + m];          a0.y = cw[(pl + 1) * 32 + m];
            a1.x = cw[pl * 32 + m + 16];     a1.y = cw[(pl + 1) * 32 + m + 16];

            const int pg0 = chunk * 32 + pl;
            const int b0 = pg0 / L;   const int l0 = pg0 - b0 * L;
            const int oh0 = l0 / WO;  const int ow0 = l0 - oh0 * WO;
            const int pg1 = pg0 + 1;
            const int b1 = pg1 / L;   const int l1 = pg1 - b1 * L;
            const int oh1 = l1 / WO;  const int ow1 = l1 - oh1 * WO;
            const float* xb0 = x + ((size_t)b0 * CIN) * HH * WW + oh0 * WW + ow0;
            const float* xb1 = x + ((size_t)b1 * CIN) * HH * WW + oh1 * WW + ow1;

#pragma unroll
            for (int kt = 0; kt < 9; ++kt) {
                const int k = kt * 16 + m;   // B column = feature index
                const int c = k / 9;  const int rr = k - c * 9;
                const int i = rr / 3; const int j = rr - i * 3;
                v2f bb;
                bb.x = xb0[(c * HH + i) * WW + j];
                bb.y = xb1[(c * HH + i) * WW + j];
                acc[0][kt] = __builtin_amdgcn_wmma_f32_16x16x4_f32(false, a0, false, bb, (short)0, acc[0][kt], false, false);
                acc[1][kt] = __builtin_amdgcn_wmma_f32_16x16x4_f32(false, a1, false, bb, (short)0, acc[1][kt], false, false);
            }
        }
        __builtin_amdgcn_wave_barrier();
    }

    // flush partial 32x144 accumulator: D row = channel, D col = feature
    float* accG = ws + WS_ACC;
#pragma unroll
    for (int t = 0; t < 2; ++t)
#pragma unroll
        for (int kt = 0; kt < 9; ++kt)
#pragma unroll
            for (int r = 0; r < 8; ++r) {
                const int ch = r + kg * 8 + t * 16;
                const int k  = m + kt * 16;
                atomicAdd(&accG[ch * KDIM + k], acc[t][kt][r]);
            }
}

// ---------------- kernel 4: delta_w = acc - (S2/S1)*w ----------------
__global__ __launch_bounds__(256) void k_fin(const float* __restrict__ w,
                                             const float* __restrict__ ws,
                                             float* __restrict__ dw) {
    const int i = blockIdx.x * blockDim.x + threadIdx.x;
    if (i >= COUT * KDIM) return;
    const int o = i / KDIM;
    float s1 = ws[WS_S1 + o];
    if (s1 == 0.f) s1 = 1.f;
    const float c = ws[WS_S2 + o] / s1;
    dw[i] = ws[WS_ACC + i] - c * w[i];
}

// ---------------- launch ----------------
extern "C" void kernel_launch(void* const* d_in, const int* in_sizes, int n_in,
                              void* d_out, int out_size, void* d_ws, size_t ws_size,
                              hipStream_t stream) {
    (void)in_sizes; (void)n_in; (void)out_size; (void)ws_size;
    const float* x    = (const float*)d_in[0];
    const float* w    = (const float*)d_in[1];
    const float* bias = (const float*)d_in[2];
    float* y  = (float*)d_out;
    float* dw = y + (size_t)P * COUT;        // delta_w after y in flat output
    float* ws = (float*)d_ws;

    k_prep<<<COUT, 64, 0, stream>>>(w, ws);
    k_fwd<<<(P / 16 + 7) / 8, 256, 0, stream>>>(x, ws, bias, y);
    k_rsum<<<504, 256, 0, stream>>>(y, ws);
    k_delta<<<1008, 256, 0, stream>>>(x, y, ws);
    k_fin<<<(COUT * KDIM + 255) / 256, 256, 0, stream>>>(w, ws, dw);
}